// Water_36472862277947
// MI455X (gfx1250) — compile-verified
//
#include <hip/hip_runtime.h>

// Water constraint kernel for MI455X (gfx1250).
// c[b,n,k] = |r_k - r_{(k+1)%3}| - l[k] over 3.2M rigid water molecules.
// Bandwidth-bound (~269 MB @ 23.3 TB/s => ~11.5us floor). Uses the CDNA5
// async global->LDS DMA path (ASYNCcnt) with a double-buffered, counted-wait
// software pipeline: compute on chunk j overlaps DMA of chunk j+1.

#define MOL_FLOATS   18     // 9 pos + 9 vel
#define MOL_BYTES    72
#define CHUNK_MOLS   32     // one wave-chunk = 32 molecules
#define CHUNK_BYTES  (CHUNK_MOLS * MOL_BYTES)     // 2304 B
#define CHUNK_VECS   (CHUNK_BYTES / 16)           // 144 x b128 transfers (5 instr/wave)
#define CHUNK_DWORDS (CHUNK_BYTES / 4)            // 576
#define NCHUNK       4      // chunks per wave (pipeline depth source)
#define WAVES_PER_BLOCK 8
#define BLOCK_MOLS   (WAVES_PER_BLOCK * NCHUNK * CHUNK_MOLS)   // 1024

__device__ __forceinline__ void water_compute_store(
    float r0, float r1, float r2, float r3, float r4, float r5,
    float r6, float r7, float r8,
    float l0, float l1, float l2, float* __restrict__ out, int m)
{
    // dr0 = p0-p1, dr1 = p1-p2, dr2 = p2-p0 (matches reference stack order)
    const float d0x = r0 - r3, d0y = r1 - r4, d0z = r2 - r5;
    const float d1x = r3 - r6, d1y = r4 - r7, d1z = r5 - r8;
    const float d2x = r6 - r0, d2y = r7 - r1, d2z = r8 - r2;

    const float c0 = sqrtf(d0x * d0x + d0y * d0y + d0z * d0z) - l0;
    const float c1 = sqrtf(d1x * d1x + d1y * d1y + d1z * d1z) - l1;
    const float c2 = sqrtf(d2x * d2x + d2y * d2y + d2z * d2z) - l2;

    const size_t o = (size_t)m * 3;
    out[o + 0] = c0;
    out[o + 1] = c1;
    out[o + 2] = c2;
}

__global__ __launch_bounds__(256) void water_main_kernel(
    const float* __restrict__ x,   // [B*N, 18]
    const float* __restrict__ l,   // [3]
    float* __restrict__ out)       // [B*N, 3]
{
    // Per-wave double buffer: 8 waves x 2 buffers x 2304 B = 36864 B.
    __shared__ float smem[WAVES_PER_BLOCK * 2 * CHUNK_DWORDS];

    const int lane = threadIdx.x & 31;
    const int wave = threadIdx.x >> 5;

    // Wave processes NCHUNK consecutive 32-molecule chunks.
    const int wave_mol0 = blockIdx.x * BLOCK_MOLS + wave * (NCHUNK * CHUNK_MOLS);

    // LDS byte address of this wave's region (low 32 bits of generic addr
    // == LDS offset per the flat->LDS aperture mapping).
    const unsigned lds_wave =
        (unsigned)(size_t)(void*)smem + (unsigned)(wave * 2 * CHUNK_BYTES);

    // Issue one chunk's DMA: 144 16-B transfers = exactly 5 async
    // instructions per wave (ASYNCcnt += 5; partial-EXEC 5th still counts 1).
    auto issue_chunk = [&](int buf, int chunk_mol) {
        const unsigned lds_b = lds_wave + (unsigned)(buf * CHUNK_BYTES);
        const unsigned g_b   = (unsigned)chunk_mol * (unsigned)MOL_BYTES;
        #pragma unroll
        for (int k = 0; k < 5; ++k) {
            const int v = k * 32 + lane;             // 0..143 (+ tail guard)
            if (v < CHUNK_VECS) {
                const unsigned lds_addr = lds_b + (unsigned)(v * 16);
                const unsigned g_off    = g_b   + (unsigned)(v * 16);
                asm volatile(
                    "global_load_async_to_lds_b128 %0, %1, %2"
                    :
                    : "v"(lds_addr), "v"(g_off), "s"(x)
                    : "memory");
            }
        }
    };

    issue_chunk(0, wave_mol0);                        // prologue: fill buf 0

    const float l0 = l[0];
    const float l1 = l[1];
    const float l2 = l[2];

    #pragma unroll
    for (int j = 0; j < NCHUNK; ++j) {
        if (j + 1 < NCHUNK) {
            // Overlap: start DMA of chunk j+1, then wait only for the OLDER
            // 5 transfers (async loads complete in order => counted wait).
            issue_chunk((j + 1) & 1, wave_mol0 + (j + 1) * CHUNK_MOLS);
            asm volatile("s_wait_asynccnt 0x5" ::: "memory");
        } else {
            asm volatile("s_wait_asynccnt 0x0" ::: "memory");
        }

        // Direct __shared__ indexing -> ds_load_*. Read stride = 18 dwords;
        // 18*lane mod 64 distinct over 32 lanes => bank-conflict-free.
        const int sbase =
            wave * (2 * CHUNK_DWORDS) + (j & 1) * CHUNK_DWORDS + lane * MOL_FLOATS;
        const float r0 = smem[sbase + 0];
        const float r1 = smem[sbase + 1];
        const float r2 = smem[sbase + 2];
        const float r3 = smem[sbase + 3];
        const float r4 = smem[sbase + 4];
        const float r5 = smem[sbase + 5];
        const float r6 = smem[sbase + 6];
        const float r7 = smem[sbase + 7];
        const float r8 = smem[sbase + 8];

        const int m = wave_mol0 + j * CHUNK_MOLS + lane;
        water_compute_store(r0, r1, r2, r3, r4, r5, r6, r7, r8,
                            l0, l1, l2, out, m);
    }
}

// Generic tail for molecules not covered by full 1024-molecule blocks
// (zero work for the 3.2M reference size; kept for safety).
__global__ __launch_bounds__(256) void water_tail_kernel(
    const float* __restrict__ x, const float* __restrict__ l,
    float* __restrict__ out, int start_mol, int total_mol)
{
    const int m = start_mol + blockIdx.x * 256 + threadIdx.x;
    if (m >= total_mol) return;
    const float* p = x + (size_t)m * MOL_FLOATS;
    water_compute_store(p[0], p[1], p[2], p[3], p[4], p[5], p[6], p[7], p[8],
                        l[0], l[1], l[2], out, m);
}

extern "C" void kernel_launch(void* const* d_in, const int* in_sizes, int n_in,
                              void* d_out, int out_size, void* d_ws, size_t ws_size,
                              hipStream_t stream) {
    const float* x = (const float*)d_in[0];   // [B, N, 18] float32
    const float* l = (const float*)d_in[1];   // [3] float32
    float* out = (float*)d_out;               // [B, N, 3] float32

    const int total_mol = in_sizes[0] / MOL_FLOATS;       // 3,200,000
    const int main_blocks = total_mol / BLOCK_MOLS;       // 3,125 (exact here)
    const int covered = main_blocks * BLOCK_MOLS;
    const int rem = total_mol - covered;

    if (main_blocks > 0) {
        hipLaunchKernelGGL(water_main_kernel,
                           dim3(main_blocks), dim3(256), 0, stream,
                           x, l, out);
    }
    if (rem > 0) {
        const int tail_blocks = (rem + 255) / 256;
        hipLaunchKernelGGL(water_tail_kernel,
                           dim3(tail_blocks), dim3(256), 0, stream,
                           x, l, out, covered, total_mol);
    }
}